// SplineEncoder_54546084659940
// MI455X (gfx1250) — compile-verified
//
#include <hip/hip_runtime.h>

typedef __attribute__((ext_vector_type(2))) float v2f;
typedef __attribute__((ext_vector_type(8))) float v8f;

#define NN 50000
#define EE 800000
#define CC 96
#define NTILE_COL (CC / 16)              // 6
#define NTILE_ROW (NN / 16)              // 3125
#define NTILES (NTILE_ROW * NTILE_COL)   // 18750
#define NC (NN * CC)                     // 4800000

// ---------------------------------------------------------------------------
// Y[N,C] = X[N,C] @ W[C,C]  (+ Y if accumulate). One wave per 16x16 tile,
// K marched in steps of 4 via V_WMMA_F32_16X16X4_F32 (full f32 precision).
// ---------------------------------------------------------------------------
__global__ __launch_bounds__(256)
void gemm_wmma_f32(const float* __restrict__ X, const float* __restrict__ W,
                   float* __restrict__ Y, int accumulate) {
    const int lane = threadIdx.x & 31;
    const int wave = threadIdx.x >> 5;
    const int tile = blockIdx.x * (blockDim.x >> 5) + wave;
    if (tile >= NTILES) return;           // wave-uniform: EXEC stays all-ones

    const int trow = tile / NTILE_COL;
    const int tcol = tile - trow * NTILE_COL;
    const int row0 = trow << 4;
    const int col0 = tcol << 4;

    const int half = lane >> 4;           // 0: lanes 0-15, 1: lanes 16-31
    const int m    = lane & 15;           // row (A) / column (B,C,D) in tile

    // C/D fragment: VGPR r holds element (M = r + 8*half, N = m)
    v8f acc;
    if (accumulate) {
#pragma unroll
        for (int r = 0; r < 8; ++r)
            acc[r] = Y[(size_t)(row0 + r + 8 * half) * CC + col0 + m];
    } else {
#pragma unroll
        for (int r = 0; r < 8; ++r) acc[r] = 0.0f;
    }

    const float* xrow = X + (size_t)(row0 + m) * CC;

#pragma unroll
    for (int k0 = 0; k0 < CC; k0 += 4) {
        const int kb = k0 + 2 * half;
        // A fragment (16x4): lanes 0-15 hold K={k0,k0+1}, lanes 16-31 K={k0+2,k0+3}
        v2f a = *(const v2f*)(xrow + kb);
        // B fragment (4x16): row K striped across lanes, mirrored half-split
        v2f b;
        b.x = W[(size_t)kb * CC + col0 + m];
        b.y = W[(size_t)(kb + 1) * CC + col0 + m];
        acc = __builtin_amdgcn_wmma_f32_16x16x4_f32(
            /*neg_a=*/false, a, /*neg_b=*/false, b,
            /*c_mod=*/(short)0, acc, /*reuse_a=*/false, /*reuse_b=*/false);
    }

#pragma unroll
    for (int r = 0; r < 8; ++r)
        Y[(size_t)(row0 + r + 8 * half) * CC + col0 + m] = acc[r];
}

// ---------------------------------------------------------------------------
// Elementwise / scatter kernels
// ---------------------------------------------------------------------------
__global__ __launch_bounds__(256)
void zero_f32(float* __restrict__ p, int n) {
    int i = blockIdx.x * blockDim.x + threadIdx.x;
    if (i < n) p[i] = 0.0f;
}

__global__ __launch_bounds__(256)
void degree_kernel(const int* __restrict__ ei, float* __restrict__ deg) {
    int e = blockIdx.x * blockDim.x + threadIdx.x;
    if (e < EE) atomicAdd(&deg[ei[EE + e]], 1.0f);
}

// agg[dst, c] += h[src, c]   (edge-major x channel-minor: coalesced)
__global__ __launch_bounds__(256)
void scatter_add(const float* __restrict__ H, const int* __restrict__ ei,
                 float* __restrict__ agg) {
    int i = blockIdx.x * blockDim.x + threadIdx.x;
    if (i >= EE * CC) return;
    int e = i / CC;
    int c = i - e * CC;
    int src = ei[e];
    int dst = ei[EE + e];
    atomicAdd(&agg[(size_t)dst * CC + c], H[(size_t)src * CC + c]);
}

// out = agg / max(deg,1) + bias   (root GEMM accumulates on top afterwards)
__global__ __launch_bounds__(256)
void combine(const float* __restrict__ agg, const float* __restrict__ deg,
             const float* __restrict__ bias, float* __restrict__ out) {
    int i = blockIdx.x * blockDim.x + threadIdx.x;
    if (i >= NC) return;
    int n = i / CC;
    int c = i - n * CC;
    out[i] = agg[i] / fmaxf(deg[n], 1.0f) + bias[c];
}

__global__ __launch_bounds__(256)
void relu_inplace(float* __restrict__ p, int n) {
    int i = blockIdx.x * blockDim.x + threadIdx.x;
    if (i < n) p[i] = fmaxf(p[i], 0.0f);
}

// ---------------------------------------------------------------------------
extern "C" void kernel_launch(void* const* d_in, const int* in_sizes, int n_in,
                              void* d_out, int out_size, void* d_ws, size_t ws_size,
                              hipStream_t stream) {
    const float* x   = (const float*)d_in[0];
    const int*   ei  = (const int*)  d_in[1];   // [2,E] int32 (JAX x64 off)
    const float* W1  = (const float*)d_in[2];
    const float* Wr1 = (const float*)d_in[3];
    const float* b1  = (const float*)d_in[4];
    const float* W2  = (const float*)d_in[5];
    const float* Wr2 = (const float*)d_in[6];
    const float* b2  = (const float*)d_in[7];
    float* out = (float*)d_out;

    float* h   = (float*)d_ws;     // [N,C]
    float* agg = h   + NC;         // [N,C]
    float* hid = agg + NC;         // [N,C]
    float* deg = hid + NC;         // [N]

    const int T = 256;
    const int gemm_blocks  = (NTILES + (T / 32) - 1) / (T / 32);
    const int nc_blocks    = (NC + T - 1) / T;
    const int edge_blocks  = (EE + T - 1) / T;
    const int scat_blocks  = (EE * CC + T - 1) / T;

    // degree (shared by both layers) + zero aggregation buffer
    zero_f32<<<(NN + T - 1) / T, T, 0, stream>>>(deg, NN);
    zero_f32<<<nc_blocks, T, 0, stream>>>(agg, NC);
    degree_kernel<<<edge_blocks, T, 0, stream>>>(ei, deg);

    // ----- layer 1: hid = relu( scatter_mean(x@W1) + x@Wr1 + b1 ) -----
    gemm_wmma_f32<<<gemm_blocks, T, 0, stream>>>(x, W1, h, 0);
    scatter_add<<<scat_blocks, T, 0, stream>>>(h, ei, agg);
    combine<<<nc_blocks, T, 0, stream>>>(agg, deg, b1, hid);
    gemm_wmma_f32<<<gemm_blocks, T, 0, stream>>>(x, Wr1, hid, 1);  // hid += x@Wr1
    relu_inplace<<<nc_blocks, T, 0, stream>>>(hid, NC);

    // ----- layer 2: out = scatter_mean(hid@W2) + hid@Wr2 + b2 -----
    zero_f32<<<nc_blocks, T, 0, stream>>>(agg, NC);
    gemm_wmma_f32<<<gemm_blocks, T, 0, stream>>>(hid, W2, h, 0);
    scatter_add<<<scat_blocks, T, 0, stream>>>(h, ei, agg);
    combine<<<nc_blocks, T, 0, stream>>>(agg, deg, b2, out);
    gemm_wmma_f32<<<gemm_blocks, T, 0, stream>>>(hid, Wr2, out, 1); // out += hid@Wr2
}